// Propagate_79164837200018
// MI455X (gfx1250) — compile-verified
//
#include <hip/hip_runtime.h>
#include <hip/hip_bf16.h>

// ---------------------------------------------------------------------------
// CDNA5 async global->LDS path (gfx1250). Builtin signature probe-confirmed:
// (AS1 v4i*, AS3 v4i*, imm offset, imm cpol).
// ---------------------------------------------------------------------------
#define AS1 __attribute__((address_space(1)))
#define AS3 __attribute__((address_space(3)))
typedef int v4i __attribute__((__vector_size__(4 * sizeof(int))));

#if defined(__has_builtin)
#if __has_builtin(__builtin_amdgcn_global_load_async_to_lds_b128) && \
    __has_builtin(__builtin_amdgcn_s_wait_asynccnt)
#define HAS_ASYNC 1
#endif
#endif
#ifndef HAS_ASYNC
#define HAS_ASYNC 0
#endif

#if HAS_ASYNC
#define ASYNC_CP_B128(gp, lp)                                                  \
  __builtin_amdgcn_global_load_async_to_lds_b128((AS1 v4i *)(gp),              \
                                                 (AS3 v4i *)(lp), 0, 0)
#define WAIT_ASYNC(n) __builtin_amdgcn_s_wait_asynccnt(n)
#endif

#if defined(__has_builtin)
#if __has_builtin(__builtin_nontemporal_load)
#define NT_LOAD(p) __builtin_nontemporal_load(p)
#endif
#endif
#ifndef NT_LOAD
#define NT_LOAD(p) (*(p))
#endif

#define FEAT 128  // F (feature dim); 32 lanes x float4 == one row per wave

// ---------------------------------------------------------------------------
// Kernel 1: zero accumulator A, per-node scalars, folded coefficient block.
// params[0]=alp*lam  params[1]=-c1  params[2]=-c2  params[3]=alp*lam0
// params[4]=1-alp*lam+c1+c2   with c_i = alp*sigmoid(lam_K[i])/K
// ---------------------------------------------------------------------------
__global__ void init_kernel(const float *__restrict__ deg0,
                            const float *__restrict__ D0,
                            const float *__restrict__ D1a,
                            const float *__restrict__ D1b,
                            const float *__restrict__ alp,
                            const float *__restrict__ lam0,
                            const float *__restrict__ lam,
                            const float *__restrict__ lam_K,
                            const int *__restrict__ Kp, float *__restrict__ A,
                            float *__restrict__ n0, float *__restrict__ r1v,
                            float *__restrict__ invden,
                            float *__restrict__ params, int N, long quarterNF) {
  long i = (long)blockIdx.x * blockDim.x + threadIdx.x;
  if (i < quarterNF) {
    ((float4 *)A)[i] = make_float4(0.f, 0.f, 0.f, 0.f);
  }
  if (i < N) {
    n0[i] = rsqrtf(deg0[i]);
    float d1 = D1a[i] + D1b[i];
    r1v[i] = rsqrtf(d1);
    invden[i] = 1.0f / (D0[i] + d1);
  }
  if (i == 0) {
    float a = alp[0], l = lam[0], l0 = lam0[0];
    float invK = 1.0f / (float)Kp[0];
    float c1 = a * (1.0f / (1.0f + expf(-lam_K[0]))) * invK;
    float c2 = a * (1.0f / (1.0f + expf(-lam_K[1]))) * invK;
    params[0] = a * l;
    params[1] = -c1;
    params[2] = -c2;
    params[3] = a * l0;
    params[4] = 1.0f - a * l + c1 + c2;
  }
}

// ---------------------------------------------------------------------------
// Kernel 2: edge scatter into one fused accumulator A.
//   grid = (Bx, 3): blockIdx.y selects the graph, so src/dst/w/f/coef are
//   bound once per block and the hot loop has no selection logic.
//   One wave per edge-row (32 lanes x float4 = 128 floats = 512 B).
//   Depth-2 pipeline: global_load_async_to_lds_b128 prefetches edge j+1's
//   Y row into LDS while edge j is scaled and scattered with non-returning
//   global_atomic_add_f32 (Y + A ~51MB -> resolves in the 192MB L2).
//   Wave-uniform loop index is forced into SGPRs via readfirstlane so the
//   per-edge src/dst/w broadcast loads can scalarize (KMcnt path).
// ---------------------------------------------------------------------------
__global__ __launch_bounds__(256) void edge_scatter(
    const float *__restrict__ Y, const int *__restrict__ src0,
    const int *__restrict__ dst0, const float *__restrict__ w0,
    const int *__restrict__ src1, const int *__restrict__ dst1,
    const float *__restrict__ w1, const int *__restrict__ src2,
    const int *__restrict__ dst2, const float *__restrict__ w2,
    const float *__restrict__ n0, const float *__restrict__ r1v,
    const float *__restrict__ params, float *__restrict__ A, int E) {
  __shared__ __align__(16) float buf[8][2][FEAT];  // 8 waves x double buffer
  const int lane = threadIdx.x & 31;
  const int wid = __builtin_amdgcn_readfirstlane(threadIdx.x >> 5);

  const int g = blockIdx.y;  // which graph / hop
  const int *S = (g == 0) ? src0 : ((g == 1) ? src1 : src2);
  const int *D = (g == 0) ? dst0 : ((g == 1) ? dst1 : dst2);
  const float *W = (g == 0) ? w0 : ((g == 1) ? w1 : w2);
  const float *f = (g == 0) ? n0 : r1v;
  const float coef = params[g];

  const int gw = __builtin_amdgcn_readfirstlane(blockIdx.x * 8 + wid);
  const int nw = gridDim.x * 8;

  auto meta = [&](int e, int &s_, int &d_, float &sc_) {
    int s = NT_LOAD(S + e);
    int d = NT_LOAD(D + e);
    float ww = NT_LOAD(W + e);
    s_ = s;
    d_ = d;
    sc_ = coef * ww * f[s] * f[d];
  };

  int srcN = 0, dstN = 0;
  float scN = 0.f;
  if (gw < E) {
    meta(gw, srcN, dstN, scN);
#if HAS_ASYNC
    ASYNC_CP_B128(Y + (long)srcN * FEAT + lane * 4, &buf[wid][0][lane * 4]);
#endif
  }
  int pb = 0;
  for (int e = gw; e < E; e += nw) {
    const int dstC = dstN;
    const float scC = scN;
#if !HAS_ASYNC
    const int srcC = srcN;
#endif
    const int en = e + nw;
    if (en < E) {
      meta(en, srcN, dstN, scN);
#if HAS_ASYNC
      ASYNC_CP_B128(Y + (long)srcN * FEAT + lane * 4,
                    &buf[wid][pb ^ 1][lane * 4]);
#endif
    }
#if HAS_ASYNC
    if (en < E) {
      WAIT_ASYNC(1);  // next copy still in flight; current buffer complete
    } else {
      WAIT_ASYNC(0);
    }
    float4 v = *(const float4 *)&buf[wid][pb][lane * 4];
#else
    float4 v = *(const float4 *)(Y + (long)srcC * FEAT + lane * 4);
#endif
    v.x *= scC; v.y *= scC; v.z *= scC; v.w *= scC;
    float *out = A + (long)dstC * FEAT + lane * 4;
    atomicAdd(out + 0, v.x);
    atomicAdd(out + 1, v.y);
    atomicAdd(out + 2, v.z);
    atomicAdd(out + 3, v.w);
    pb ^= 1;
  }
}

// ---------------------------------------------------------------------------
// Kernel 3: Y_new = sY*Y - alp*lam0*(Y-X)*invden[node] + A   (float4)
// ---------------------------------------------------------------------------
__global__ void finalize_kernel(const float *__restrict__ Y,
                                const float *__restrict__ X,
                                const float *__restrict__ A,
                                const float *__restrict__ invden,
                                const float *__restrict__ params,
                                float *__restrict__ out, int N, int Fq) {
  long i = (long)blockIdx.x * blockDim.x + threadIdx.x;
  long T = (long)N * Fq;
  if (i >= T) return;
  int node = (int)((unsigned long)i / (unsigned)Fq);
  float al0 = params[3];
  float sy = params[4];
  float inv = invden[node];
  float4 y = ((const float4 *)Y)[i];
  float4 x = ((const float4 *)X)[i];
  float4 a = ((const float4 *)A)[i];
  float4 o;
  o.x = sy * y.x - al0 * (y.x - x.x) * inv + a.x;
  o.y = sy * y.y - al0 * (y.y - x.y) * inv + a.y;
  o.z = sy * y.z - al0 * (y.z - x.z) * inv + a.z;
  o.w = sy * y.w - al0 * (y.w - x.w) * inv + a.w;
  ((float4 *)out)[i] = o;
}

// ---------------------------------------------------------------------------
extern "C" void kernel_launch(void *const *d_in, const int *in_sizes, int n_in,
                              void *d_out, int out_size, void *d_ws,
                              size_t ws_size, hipStream_t stream) {
  const float *Y = (const float *)d_in[0];
  const float *X = (const float *)d_in[1];
  const float *alp = (const float *)d_in[2];
  const float *lam0 = (const float *)d_in[3];
  const float *lam = (const float *)d_in[4];
  const float *lam_K = (const float *)d_in[5];
  const int *src0 = (const int *)d_in[6];
  const int *dst0 = (const int *)d_in[7];
  const float *w0 = (const float *)d_in[8];
  const float *deg0 = (const float *)d_in[9];
  const int *src1 = (const int *)d_in[10];
  const int *dst1 = (const int *)d_in[11];
  const float *w1 = (const float *)d_in[12];
  const int *src2 = (const int *)d_in[13];
  const int *dst2 = (const int *)d_in[14];
  const float *w2 = (const float *)d_in[15];
  const float *D0 = (const float *)d_in[16];
  const float *D1a = (const float *)d_in[17];
  const float *D1b = (const float *)d_in[18];
  const int *Kp = (const int *)d_in[19];

  const int N = in_sizes[9];          // deg0 length
  const int E = in_sizes[6];          // edges per graph
  const int F = in_sizes[0] / N;      // 128

  // workspace layout (floats): A[N*F] | n0[N] | r1[N] | invden[N] | params[8]
  float *ws = (float *)d_ws;
  float *A = ws;
  float *n0 = ws + (size_t)N * F;
  float *r1v = n0 + N;
  float *invden = r1v + N;
  float *params = invden + N;

  const long quarterNF = (long)N * F / 4;
  const long maxT = (quarterNF > N) ? quarterNF : N;

  init_kernel<<<(int)((maxT + 255) / 256), 256, 0, stream>>>(
      deg0, D0, D1a, D1b, alp, lam0, lam, lam_K, Kp, A, n0, r1v, invden,
      params, N, quarterNF);

  edge_scatter<<<dim3(2048, 3, 1), 256, 0, stream>>>(
      Y, src0, dst0, w0, src1, dst1, w1, src2, dst2, w2, n0, r1v, params, A,
      E);

  finalize_kernel<<<(int)((quarterNF + 255) / 256), 256, 0, stream>>>(
      Y, X, A, invden, params, (float *)d_out, N, F / 4);
}